// GlobalAttentionSeq2Seq_21818433863710
// MI455X (gfx1250) — compile-verified
//
#include <hip/hip_runtime.h>
#include <math.h>

#define BATCH 64
#define TE 128
#define TD 64
#define EMB 4
#define HID 8
#define VOC 32000
#define MROWS (BATCH*TD)   // 4096
#define RROWS 8            // rows per block in the logsumexp pass

typedef __attribute__((ext_vector_type(2))) float v2f;
typedef __attribute__((ext_vector_type(8))) float v8f;

__device__ __forceinline__ float sigmoidf_(float x) { return 1.0f / (1.0f + expf(-x)); }

// ---------------- Encoder GRU (one block per batch element) ----------------
__global__ __launch_bounds__(128) void enc_kernel(
    const int* __restrict__ enc_input,   // [64,128]
    const float* __restrict__ embed,     // [32000,4]
    const float* __restrict__ w_ih,      // [24,4]
    const float* __restrict__ w_hh,      // [24,8]
    const float* __restrict__ b_ih,      // [24]
    const float* __restrict__ b_hh,      // [24]
    const float* __restrict__ e2d_w,     // [8,8]
    const float* __restrict__ e2d_b,     // [8]
    float* __restrict__ enc_out,         // [64,128,8]
    float* __restrict__ hdec)            // [64,8]
{
    __shared__ float gi[TE * 24];
    __shared__ float h[HID];
    __shared__ float hn[HID];
    const int b = blockIdx.x;
    const int tid = threadIdx.x;

    // precompute all input-side gates: gi[t][g] = emb(t) . w_ih[g] + b_ih[g]
    for (int idx = tid; idx < TE * 24; idx += blockDim.x) {
        const int t = idx / 24, g = idx % 24;
        const int tok = enc_input[b * TE + t];
        const float* e = embed + tok * EMB;
        const float* w = w_ih + g * EMB;
        float acc = b_ih[g];
#pragma unroll
        for (int k = 0; k < EMB; ++k) acc += e[k] * w[k];
        gi[idx] = acc;
    }
    if (tid < HID) h[tid] = 0.0f;
    __syncthreads();

    // recurrent weights for thread j (<8)
    float whr[HID] = {}, whz[HID] = {}, whn[HID] = {};
    float bhr = 0.f, bhz = 0.f, bhn = 0.f;
    if (tid < HID) {
#pragma unroll
        for (int k = 0; k < HID; ++k) {
            whr[k] = w_hh[tid * HID + k];
            whz[k] = w_hh[(HID + tid) * HID + k];
            whn[k] = w_hh[(2 * HID + tid) * HID + k];
        }
        bhr = b_hh[tid]; bhz = b_hh[HID + tid]; bhn = b_hh[2 * HID + tid];
    }

    for (int t = 0; t < TE; ++t) {
        if (tid < HID) {
            float ghr = bhr, ghz = bhz, ghn = bhn;
#pragma unroll
            for (int k = 0; k < HID; ++k) {
                const float hk = h[k];
                ghr += hk * whr[k]; ghz += hk * whz[k]; ghn += hk * whn[k];
            }
            const float gir = gi[t * 24 + tid];
            const float giz = gi[t * 24 + HID + tid];
            const float gin = gi[t * 24 + 2 * HID + tid];
            const float r = sigmoidf_(gir + ghr);
            const float z = sigmoidf_(giz + ghz);
            const float n = tanhf(gin + r * ghn);
            const float hv = (1.0f - z) * n + z * h[tid];
            hn[tid] = hv;
            enc_out[(b * TE + t) * HID + tid] = hv;
        }
        __syncthreads();
        if (tid < HID) h[tid] = hn[tid];
        __syncthreads();
    }
    if (tid < HID) {
        float acc = e2d_b[tid];
#pragma unroll
        for (int k = 0; k < HID; ++k) acc += h[k] * e2d_w[tid * HID + k];
        hdec[b * HID + tid] = acc;
    }
}

// ---------------- Decoder attention + GRUCell (one block per batch) ----------------
__global__ __launch_bounds__(128) void dec_kernel(
    const int* __restrict__ dec_input,   // [64,64]
    const float* __restrict__ embed,     // [32000,4]
    const float* __restrict__ att_w,     // [8,8]
    const float* __restrict__ cw_ih,     // [24,12]
    const float* __restrict__ cw_hh,     // [24,8]
    const float* __restrict__ cb_ih,     // [24]
    const float* __restrict__ cb_hh,     // [24]
    const float* __restrict__ enc_out,   // [64,128,8]
    const float* __restrict__ hdec,      // [64,8]
    float* __restrict__ dec_out)         // [4096,8]
{
    __shared__ float enc[TE * HID];
    __shared__ float sc[TE];
    __shared__ float red[TE];
    __shared__ float h[HID], hn[HID], q[HID], ctx[HID], et[EMB];
    const int b = blockIdx.x;
    const int tid = threadIdx.x;

    for (int i = tid; i < TE * HID; i += blockDim.x) enc[i] = enc_out[b * TE * HID + i];
    if (tid < HID) h[tid] = hdec[b * HID + tid];
    __syncthreads();

    float aw[HID] = {};
    float wir[EMB + HID] = {}, wiz[EMB + HID] = {}, win[EMB + HID] = {};
    float whr[HID] = {}, whz[HID] = {}, whn[HID] = {};
    float bir = 0.f, biz = 0.f, bin_ = 0.f, bhr = 0.f, bhz = 0.f, bhn = 0.f;
    if (tid < HID) {
#pragma unroll
        for (int k = 0; k < HID; ++k) aw[k] = att_w[tid * HID + k];
#pragma unroll
        for (int k = 0; k < EMB + HID; ++k) {
            wir[k] = cw_ih[tid * (EMB + HID) + k];
            wiz[k] = cw_ih[(HID + tid) * (EMB + HID) + k];
            win[k] = cw_ih[(2 * HID + tid) * (EMB + HID) + k];
        }
#pragma unroll
        for (int k = 0; k < HID; ++k) {
            whr[k] = cw_hh[tid * HID + k];
            whz[k] = cw_hh[(HID + tid) * HID + k];
            whn[k] = cw_hh[(2 * HID + tid) * HID + k];
        }
        bir = cb_ih[tid]; biz = cb_ih[HID + tid]; bin_ = cb_ih[2 * HID + tid];
        bhr = cb_hh[tid]; bhz = cb_hh[HID + tid]; bhn = cb_hh[2 * HID + tid];
    }

    for (int t = 0; t < TD; ++t) {
        if (tid < EMB) {
            const int tok = dec_input[b * TD + t];
            et[tid] = embed[tok * EMB + tid];
        }
        if (tid < HID) {
            float acc = 0.0f;
#pragma unroll
            for (int k = 0; k < HID; ++k) acc += h[k] * aw[k];
            q[tid] = acc;
        }
        __syncthreads();
        // attention scores (one timestep per thread)
        {
            float s = 0.0f;
#pragma unroll
            for (int k = 0; k < HID; ++k) s += enc[tid * HID + k] * q[k];
            sc[tid] = s; red[tid] = s;
        }
        __syncthreads();
        for (int s = TE / 2; s > 0; s >>= 1) {   // max reduce
            if (tid < s) red[tid] = fmaxf(red[tid], red[tid + s]);
            __syncthreads();
        }
        const float m = red[0];
        __syncthreads();
        const float p = expf(sc[tid] - m);
        sc[tid] = p; red[tid] = p;
        __syncthreads();
        for (int s = TE / 2; s > 0; s >>= 1) {   // sum reduce
            if (tid < s) red[tid] += red[tid + s];
            __syncthreads();
        }
        const float ssum = red[0];
        if (tid < HID) {
            float c = 0.0f;
            for (int k = 0; k < TE; ++k) c += sc[k] * enc[k * HID + tid];
            ctx[tid] = c / ssum;
        }
        __syncthreads();
        if (tid < HID) {
            float gir = bir, giz = biz, gin = bin_;
#pragma unroll
            for (int k = 0; k < EMB; ++k) {
                const float x = et[k];
                gir += x * wir[k]; giz += x * wiz[k]; gin += x * win[k];
            }
#pragma unroll
            for (int k = 0; k < HID; ++k) {
                const float x = ctx[k];
                gir += x * wir[EMB + k]; giz += x * wiz[EMB + k]; gin += x * win[EMB + k];
            }
            float ghr = bhr, ghz = bhz, ghn = bhn;
#pragma unroll
            for (int k = 0; k < HID; ++k) {
                const float hk = h[k];
                ghr += hk * whr[k]; ghz += hk * whz[k]; ghn += hk * whn[k];
            }
            const float r = sigmoidf_(gir + ghr);
            const float z = sigmoidf_(giz + ghz);
            const float n = tanhf(gin + r * ghn);
            const float hv = (1.0f - z) * n + z * h[tid];
            hn[tid] = hv;
            dec_out[(b * TD + t) * HID + tid] = hv;
        }
        __syncthreads();
        if (tid < HID) h[tid] = hn[tid];
        __syncthreads();
    }
}

// ---- Pass 1: per-row online logsumexp over the vocab (8 rows / block) ----
__global__ __launch_bounds__(256) void rowred_kernel(
    const float* __restrict__ dec_out,  // [4096,8]
    const float* __restrict__ out_w,    // [32000,8]
    const float* __restrict__ out_b,    // [32000]
    float* __restrict__ rls)            // [4096]  max + log(sum exp)
{
    __shared__ float sm[RROWS][256];
    __shared__ float ss[RROWS][256];
    const int rb  = blockIdx.x * RROWS;
    const int tid = threadIdx.x;

    // 8 rows' hidden states in registers (fully unrolled constant indexing)
    float hh[RROWS][HID];
#pragma unroll
    for (int r = 0; r < RROWS; ++r)
#pragma unroll
        for (int k = 0; k < HID; ++k)
            hh[r][k] = dec_out[(rb + r) * HID + k];

    float m[RROWS], s[RROWS];
#pragma unroll
    for (int r = 0; r < RROWS; ++r) { m[r] = -INFINITY; s[r] = 0.0f; }

    for (int v = tid; v < VOC; v += 256) {
        const float* w = out_w + v * HID;
        float wv[HID];
#pragma unroll
        for (int k = 0; k < HID; ++k) wv[k] = w[k];
        const float bv = out_b[v];
#pragma unroll
        for (int r = 0; r < RROWS; ++r) {
            float x = bv;
#pragma unroll
            for (int k = 0; k < HID; ++k) x += hh[r][k] * wv[k];
            if (x > m[r]) { s[r] = s[r] * expf(m[r] - x) + 1.0f; m[r] = x; }
            else          { s[r] += expf(x - m[r]); }
        }
    }
#pragma unroll
    for (int r = 0; r < RROWS; ++r) { sm[r][tid] = m[r]; ss[r][tid] = s[r]; }
    __syncthreads();

    // one wave per row: serial-merge 8 strided entries, then shfl tree
    const int row  = tid >> 5;          // 8 waves == RROWS
    const int lane = tid & 31;
    float mm = sm[row][lane], sacc = ss[row][lane];
    for (int i = lane + 32; i < 256; i += 32) {
        const float m2 = sm[row][i], s2 = ss[row][i];
        const float mx = fmaxf(mm, m2);
        sacc = sacc * expf(mm - mx) + s2 * expf(m2 - mx);
        mm = mx;
    }
#pragma unroll
    for (int off = 16; off > 0; off >>= 1) {
        const float m2 = __shfl_down(mm, off);
        const float s2 = __shfl_down(sacc, off);
        const float mx = fmaxf(mm, m2);
        sacc = sacc * expf(mm - mx) + s2 * expf(m2 - mx);
        mm = mx;
    }
    if (lane == 0) rls[rb + row] = mm + logf(sacc);
}

// ---- Pass 2: WMMA f32 GEMM tile + fused bias/log_softmax, NT stores ----
__global__ __launch_bounds__(256) void vocab_kernel(
    const float* __restrict__ dec_out,  // A: [4096,8]
    const float* __restrict__ out_w,    // B source: [32000,8]
    const float* __restrict__ out_b,    // [32000]
    const float* __restrict__ rls,      // [4096]
    float* __restrict__ out)            // [4096,32000]
{
    const int tid  = threadIdx.x;
    const int lane = tid & 31;
    const int wv   = tid >> 5;                      // 8 waves / block
    const int tile = blockIdx.x * 8 + wv;           // 256*2000 tiles, divides evenly
    const int nt   = tile % (VOC / 16);
    const int mt   = tile / (VOC / 16);

    const int half = lane >> 4;                     // 0: K=0,1  1: K=2,3
    const int l    = lane & 15;
    const int k0   = half * 2;

    // A 16x4 f32 layout: lanes 0-15 VGPR0=K0,VGPR1=K1; lanes 16-31 K2,K3 (M = l)
    const int m = mt * 16 + l;
    // B 4x16 f32 layout mirrors A with N = l
    const int n = nt * 16 + l;

    v2f a0, a1, b0, b1;
    a0.x = dec_out[m * HID + k0];     a0.y = dec_out[m * HID + k0 + 1];
    a1.x = dec_out[m * HID + k0 + 4]; a1.y = dec_out[m * HID + k0 + 5];
    b0.x = out_w[n * HID + k0];       b0.y = out_w[n * HID + k0 + 1];
    b1.x = out_w[n * HID + k0 + 4];   b1.y = out_w[n * HID + k0 + 5];

    v8f c = {0.f, 0.f, 0.f, 0.f, 0.f, 0.f, 0.f, 0.f};
    c = __builtin_amdgcn_wmma_f32_16x16x4_f32(false, a0, false, b0, (short)0, c, false, false);
    c = __builtin_amdgcn_wmma_f32_16x16x4_f32(false, a1, false, b1, (short)0, c, false, false);

    // D layout: VGPR r -> row (mt*16 + half*8 + r), col = nt*16 + l
    const float bias = out_b[n];
    const int rbase = mt * 16 + half * 8;
#pragma unroll
    for (int r = 0; r < 8; ++r) {
        const int row = rbase + r;
        __builtin_nontemporal_store(c[r] + bias - rls[row],
                                    out + (size_t)row * VOC + n);
    }
}

extern "C" void kernel_launch(void* const* d_in, const int* in_sizes, int n_in,
                              void* d_out, int out_size, void* d_ws, size_t ws_size,
                              hipStream_t stream) {
    const int*   enc_input = (const int*)d_in[0];
    const int*   dec_input = (const int*)d_in[1];
    const float* embed     = (const float*)d_in[2];
    const float* gru_w_ih  = (const float*)d_in[3];
    const float* gru_w_hh  = (const float*)d_in[4];
    const float* gru_b_ih  = (const float*)d_in[5];
    const float* gru_b_hh  = (const float*)d_in[6];
    const float* att_w     = (const float*)d_in[7];
    const float* e2d_w     = (const float*)d_in[8];
    const float* e2d_b     = (const float*)d_in[9];
    const float* cw_ih     = (const float*)d_in[10];
    const float* cw_hh     = (const float*)d_in[11];
    const float* cb_ih     = (const float*)d_in[12];
    const float* cb_hh     = (const float*)d_in[13];
    const float* out_w     = (const float*)d_in[14];
    const float* out_b     = (const float*)d_in[15];
    float* out = (float*)d_out;

    float* ws      = (float*)d_ws;
    float* enc_out = ws;                                // 64*128*8 = 65536 floats
    float* hdec    = enc_out + BATCH * TE * HID;        // 512
    float* dec_out = hdec + BATCH * HID;                // 4096*8 = 32768
    float* rls     = dec_out + MROWS * HID;             // 4096

    enc_kernel<<<BATCH, 128, 0, stream>>>(enc_input, embed, gru_w_ih, gru_w_hh,
                                          gru_b_ih, gru_b_hh, e2d_w, e2d_b,
                                          enc_out, hdec);
    dec_kernel<<<BATCH, 128, 0, stream>>>(dec_input, embed, att_w, cw_ih, cw_hh,
                                          cb_ih, cb_hh, enc_out, hdec, dec_out);
    rowred_kernel<<<MROWS / RROWS, 256, 0, stream>>>(dec_out, out_w, out_b, rls);
    vocab_kernel<<<(MROWS / 16) * (VOC / 16) / 8, 256, 0, stream>>>(dec_out, out_w,
                                                                    out_b, rls, out);
}